// Conv1DfftAutograd_69174743270240
// MI455X (gfx1250) — compile-verified
//
#include <hip/hip_runtime.h>
#include <cstddef>
#include <cstdint>

// ============================================================================
// FFT-based 1D conv with spectral truncation, MI455X (gfx1250, wave32).
//
// Pipeline (all fp32, bandwidth-bound; ~5 GB total HBM traffic @ 23.3 TB/s):
//   1) k_twiddle  : e^{-2*pi*i*j/L} table, L = 131072
//   2) k_load     : zero-pad x (256 sigs) and w (128 sigs) into complex buf
//   3) k_fft_high + k_fft_low (DIF) : forward FFT -> bit-reversed slot order
//   4) k_einsum   : per slot, 32x16x16 real matmuls via v_wmma_f32_16x16x4_f32
//                   (truncation mask via bin = brev17(slot); Hermitian mirror
//                    bins computed directly, so irfft assembly is free)
//   5) k_fft_low + k_fft_high (DIT, conj twiddles) : inverse FFT (bitrev in,
//                   natural out)
//   6) k_extract  : real part * 1/L + bias, crop to out_W
//
// Workspace layout (assumes ws_size >= ~900 MB):
//   tw   : 65536 float2          (0.5 MB)
//   Xc   : 384 * 131072 float2   (384 MB)   forward spectra (x then w)
//   Oc   : 512 * 131072 float2   (512 MB)   product spectra / inverse buffer
// ============================================================================

constexpr int kN = 32, kC = 8, kW = 65536;
constexpr int kF = 16, kKer = 128;
constexpr int kOutW = kW - kKer + 1;            // 65409
constexpr int kL = 131072;                      // FFT length
constexpr int kLog2L = 17;
constexpr int kHalfTw = kL / 2;                 // 65536 twiddle entries
constexpr int kKeep = kL / 2 + 1 - 32768;       // 32769 kept rfft bins
constexpr int kSigX = kN * kC;                  // 256
constexpr int kSigFwd = kSigX + kF * kC;        // 384
constexpr int kSigOut = kN * kF;                // 512

typedef float v2f __attribute__((ext_vector_type(2)));
typedef float v8f __attribute__((ext_vector_type(8)));

__device__ __forceinline__ float2 cmul(float2 a, float2 b) {
  return make_float2(a.x * b.x - a.y * b.y, a.x * b.y + a.y * b.x);
}
__device__ __forceinline__ float pick(float2 v, int hi) { return hi ? v.y : v.x; }

// ---------------------------------------------------------------------------
__global__ void k_twiddle(float2* __restrict__ tw) {
  int j = blockIdx.x * blockDim.x + threadIdx.x;
  if (j < kHalfTw) {
    double ang = -6.283185307179586476925286766559 * (double)j / (double)kL;
    tw[j] = make_float2((float)cos(ang), (float)sin(ang));
  }
}

// Zero-pad real inputs into complex FFT buffer (natural order).
__global__ void k_load(const float* __restrict__ x, const float* __restrict__ w,
                       float2* __restrict__ Xc) {
  size_t gid = (size_t)blockIdx.x * 256 + threadIdx.x;
  int s = (int)(gid >> kLog2L);
  int j = (int)(gid & (kL - 1));
  float v = 0.0f;
  if (s < kSigX) {
    if (j < kW) {
      v = x[((size_t)s << 16) + j];
      if (j + 4096 < kW) __builtin_prefetch(&x[((size_t)s << 16) + j + 4096], 0, 0);
    }
  } else {
    int si = s - kSigX;
    if (j < kKer) v = w[si * kKer + j];
  }
  Xc[gid] = make_float2(v, 0.0f);
}

// Spans 2^17 .. 2^12 (strided sets {i*2048 + r}); fwd = DIF first, inv = DIT last.
__global__ void k_fft_high(float2* __restrict__ buf, const float2* __restrict__ twd,
                           int inverse) {
  __shared__ float2 lds[64 * 33];  // pitch 33 to dodge bank conflicts
  const size_t sigbase = (size_t)blockIdx.y * kL;
  const int r0 = blockIdx.x * 32;

  for (int t = threadIdx.x; t < 2048; t += 256) {
    int i = t >> 5, r = t & 31;
    lds[i * 33 + r] = buf[sigbase + (size_t)i * 2048 + (r0 + r)];
  }
  __syncthreads();

  for (int si = 0; si < 6; ++si) {
    int s_exp = inverse ? (12 + si) : (17 - si);
    int half_i = 1 << (s_exp - 12);
#pragma unroll
    for (int b = 0; b < 4; ++b) {
      int idx = threadIdx.x + (b << 8);
      int r = idx & 31;
      int ib = idx >> 5;
      int p_i = ib & (half_i - 1);
      int lo_i = ((ib >> (s_exp - 12)) << (s_exp - 11)) + p_i;
      int hi_i = lo_i + half_i;
      int p = (p_i << 11) + r0 + r;
      float2 wv = twd[p << (kLog2L - s_exp)];
      if (inverse) wv.y = -wv.y;
      float2 u = lds[lo_i * 33 + r];
      float2 t2 = lds[hi_i * 33 + r];
      if (inverse) {  // DIT
        float2 vv = cmul(t2, wv);
        lds[lo_i * 33 + r] = make_float2(u.x + vv.x, u.y + vv.y);
        lds[hi_i * 33 + r] = make_float2(u.x - vv.x, u.y - vv.y);
      } else {        // DIF
        lds[lo_i * 33 + r] = make_float2(u.x + t2.x, u.y + t2.y);
        lds[hi_i * 33 + r] = cmul(make_float2(u.x - t2.x, u.y - t2.y), wv);
      }
    }
    __syncthreads();
  }

  for (int t = threadIdx.x; t < 2048; t += 256) {
    int i = t >> 5, r = t & 31;
    buf[sigbase + (size_t)i * 2048 + (r0 + r)] = lds[i * 33 + r];
  }
}

// Spans 2^11 .. 2 on contiguous 2048-element blocks.
__global__ void k_fft_low(float2* __restrict__ buf, const float2* __restrict__ twd,
                          int inverse) {
  __shared__ float2 lds[2048];
  const size_t base = (size_t)blockIdx.y * kL + (size_t)blockIdx.x * 2048;

  for (int t = threadIdx.x; t < 2048; t += 256) lds[t] = buf[base + t];
  __syncthreads();

  for (int si = 0; si < 11; ++si) {
    int s_exp = inverse ? (si + 1) : (11 - si);
    int half = 1 << (s_exp - 1);
#pragma unroll
    for (int b = 0; b < 4; ++b) {
      int idx = threadIdx.x + (b << 8);
      int p = idx & (half - 1);
      int lo = ((idx >> (s_exp - 1)) << s_exp) + p;
      int hi = lo + half;
      float2 wv = twd[p << (kLog2L - s_exp)];
      if (inverse) wv.y = -wv.y;
      float2 u = lds[lo];
      float2 t2 = lds[hi];
      if (inverse) {
        float2 vv = cmul(t2, wv);
        lds[lo] = make_float2(u.x + vv.x, u.y + vv.y);
        lds[hi] = make_float2(u.x - vv.x, u.y - vv.y);
      } else {
        lds[lo] = make_float2(u.x + t2.x, u.y + t2.y);
        lds[hi] = cmul(make_float2(u.x - t2.x, u.y - t2.y), wv);
      }
    }
    __syncthreads();
  }

  for (int t = threadIdx.x; t < 2048; t += 256) buf[base + t] = lds[t];
}

// Per-slot complex einsum via fp32 WMMA. One wave per slot (8 slots/block).
//   out_re[n,f] = [xr|xi](32x16) x [yr;yi](16x16)
//   out_im[n,f] = [xr|xi](32x16) x [-yi;yr](16x16)
// K decomposed into 4 steps of v_wmma_f32_16x16x4_f32; M=32 -> 2 tiles.
// A frag: lane (h=lane/16): rows M=lane%16, K = 4q + 2h + v (v = vgpr 0/1)
// B frag: cols N=lane%16,  K = 4q + 2h + v   (same K formula)
__global__ void k_einsum(const float2* __restrict__ Xc, const float2* __restrict__ Wc,
                         float2* __restrict__ Oc) {
  __shared__ float2 tile[512][8];  // [n*16+f][slot-in-block], 32 KB
  const int lane = threadIdx.x & 31;
  const int wave = threadIdx.x >> 5;
  const int j0 = blockIdx.x * 8;
  const int j = j0 + wave;                       // slot (bit-reversed bin order)
  const unsigned bin = __brev((unsigned)j) >> (32 - kLog2L);
  const bool kept = (bin < (unsigned)kKeep) || (bin > (unsigned)(kL - kKeep));

  const int fcol = lane & 15;  // N index (f) / M row within tile
  const int h = lane >> 4;

  v8f accRe0 = {0, 0, 0, 0, 0, 0, 0, 0}, accIm0 = {0, 0, 0, 0, 0, 0, 0, 0};
  v8f accRe1 = {0, 0, 0, 0, 0, 0, 0, 0}, accIm1 = {0, 0, 0, 0, 0, 0, 0, 0};

  if (kept) {  // uniform per wave -> EXEC stays all-ones inside
    // Per-lane complex loads: c = a*4 + 2h + v covers all K with re/im split.
    float2 Y[2][2], XA[2][2][2];
#pragma unroll
    for (int a = 0; a < 2; ++a)
#pragma unroll
      for (int v = 0; v < 2; ++v) {
        int c = a * 4 + 2 * h + v;
        Y[a][v] = Wc[(size_t)(fcol * kC + c) * kL + j];
        XA[0][a][v] = Xc[(size_t)((fcol)*kC + c) * kL + j];
        XA[1][a][v] = Xc[(size_t)((16 + fcol) * kC + c) * kL + j];
      }
#pragma unroll
    for (int q = 0; q < 4; ++q) {
      const int a = q & 1;
      const int hi = q >> 1;            // K>=8 half -> imag component
      const float sgn = (q < 2) ? -1.0f : 1.0f;
      v2f b1, b2, af0, af1;
      b1[0] = pick(Y[a][0], hi);        b1[1] = pick(Y[a][1], hi);
      b2[0] = sgn * pick(Y[a][0], hi ^ 1);
      b2[1] = sgn * pick(Y[a][1], hi ^ 1);
      af0[0] = pick(XA[0][a][0], hi);   af0[1] = pick(XA[0][a][1], hi);
      af1[0] = pick(XA[1][a][0], hi);   af1[1] = pick(XA[1][a][1], hi);
#if defined(__HIP_DEVICE_COMPILE__)
      accRe0 = __builtin_amdgcn_wmma_f32_16x16x4_f32(false, af0, false, b1,
                                                     (short)0, accRe0, false, false);
      accIm0 = __builtin_amdgcn_wmma_f32_16x16x4_f32(false, af0, false, b2,
                                                     (short)0, accIm0, false, false);
      accRe1 = __builtin_amdgcn_wmma_f32_16x16x4_f32(false, af1, false, b1,
                                                     (short)0, accRe1, false, false);
      accIm1 = __builtin_amdgcn_wmma_f32_16x16x4_f32(false, af1, false, b2,
                                                     (short)0, accIm1, false, false);
#else
      // Host pass never emits device code; keep variables formally used.
      (void)b1; (void)b2; (void)af0; (void)af1;
#endif
    }
  }

  // D layout: vgpr d -> M = d + 8h, N = lane%16. Stage through LDS so the
  // per-signal global stores become 64B-contiguous chunks.
#pragma unroll
  for (int d = 0; d < 8; ++d) {
    int n0 = d + 8 * h;
    tile[n0 * kF + fcol][wave] = make_float2(accRe0[d], accIm0[d]);
    tile[(16 + n0) * kF + fcol][wave] = make_float2(accRe1[d], accIm1[d]);
  }
  __syncthreads();

#pragma unroll
  for (int sbase = 0; sbase < 512; sbase += 256) {
    int sig = sbase + threadIdx.x;
    float4* dst = (float4*)(Oc + (size_t)sig * kL + j0);
    const float2* src = tile[sig];
#pragma unroll
    for (int i = 0; i < 4; ++i) {
      float2 e0 = src[2 * i], e1 = src[2 * i + 1];
      dst[i] = make_float4(e0.x, e0.y, e1.x, e1.y);
    }
  }
}

// Real part * 1/L + bias, cropped to out_W.
__global__ void k_extract(const float2* __restrict__ Oc, const float* __restrict__ bias,
                          float* __restrict__ out) {
  size_t gid = (size_t)blockIdx.x * 256 + threadIdx.x;
  if (gid >= (size_t)kSigOut * kOutW) return;
  int sig = (int)(gid / kOutW);
  int t = (int)(gid - (size_t)sig * kOutW);
  int f = sig & (kF - 1);
  out[gid] = Oc[(size_t)sig * kL + t].x * (1.0f / (float)kL) + bias[f];
}

// ---------------------------------------------------------------------------
extern "C" void kernel_launch(void* const* d_in, const int* in_sizes, int n_in,
                              void* d_out, int out_size, void* d_ws, size_t ws_size,
                              hipStream_t stream) {
  (void)in_sizes; (void)n_in; (void)out_size; (void)ws_size;
  const float* x = (const float*)d_in[0];
  const float* w = (const float*)d_in[1];
  const float* bias = (const float*)d_in[2];
  float* out = (float*)d_out;

  float2* tw = (float2*)d_ws;
  float2* Xc = tw + kHalfTw;
  float2* Oc = Xc + (size_t)kSigFwd * kL;

  k_twiddle<<<kHalfTw / 256, 256, 0, stream>>>(tw);
  k_load<<<(unsigned)(((size_t)kSigFwd * kL) / 256), 256, 0, stream>>>(x, w, Xc);
  // forward DIF: big spans first
  k_fft_high<<<dim3(64, kSigFwd), 256, 0, stream>>>(Xc, tw, 0);
  k_fft_low<<<dim3(64, kSigFwd), 256, 0, stream>>>(Xc, tw, 0);
  // per-bin complex matmul (WMMA), truncation mask applied in bitrev order
  k_einsum<<<kL / 8, 256, 0, stream>>>(Xc, Xc + (size_t)kSigX * kL, Oc);
  // inverse DIT: small spans first, conj twiddles
  k_fft_low<<<dim3(64, kSigOut), 256, 0, stream>>>(Oc, tw, 1);
  k_fft_high<<<dim3(64, kSigOut), 256, 0, stream>>>(Oc, tw, 1);
  size_t tot = (size_t)kSigOut * kOutW;
  k_extract<<<(unsigned)((tot + 255) / 256), 256, 0, stream>>>(Oc, bias, out);
}